// BeamConv1D_82566451298958
// MI455X (gfx1250) — compile-verified
//
#include <hip/hip_runtime.h>
#include <stdint.h>

typedef float    v2f __attribute__((ext_vector_type(2)));
typedef float    v8f __attribute__((ext_vector_type(8)));
typedef uint32_t v4u __attribute__((ext_vector_type(4)));
typedef uint32_t v8u __attribute__((ext_vector_type(8)));

#define NPTS   500000
#define CIN    64
#define FOUT   64
#define KW     3
#define NSEGS  65536
#define TILE   256            // points per block
#define XSTR   68             // LDS row stride (64 + 4 pad): 16B-aligned rows, bank-spread
#define XROWS  (TILE + 2)     // halo rows for K=3 'SAME' conv
#define SMEMB  ((XROWS * XSTR + KW * CIN * FOUT) * 4)

// Fused conv kernel:  out[p,f] = bias[f] + sum_{k,c} srcrow(p+k-1)[c] * w[k][c][f]
//   GATHER=1:  srcrow(i) = src[inv[i]]   (pre-conv on gathered X)
//   GATHER=0:  srcrow(i) = src[i]        (post-conv on segment buffer)
//   SCATTER=1: atomicAdd into dst[segid[p]]   (segment_sum fusion)
//   SCATTER=0: dst[p] = value
template <bool GATHER, bool SCATTER>
__global__ __launch_bounds__(256) void beam_conv_wmma(
    const float* __restrict__ src, const float* __restrict__ w,
    const float* __restrict__ bias, const int* __restrict__ inv,
    const int* __restrict__ segid, float* __restrict__ dst, int L)
{
    extern __shared__ float smem[];
    float* xs  = smem;                   // [XROWS][XSTR] staged input rows
    float* wsm = smem + XROWS * XSTR;    // [KW][CIN][FOUT] weights

    const int tid = threadIdx.x;
    const int n0  = blockIdx.x * TILE;

    // ---- stage weights via Tensor Data Mover: one DMA, issued by wave 0 only ----
    if (tid < 32) {
        const uint64_t gaddr = (uint64_t)(uintptr_t)w;
        v4u g0;
        g0[0] = 1u;                                   // count=1, load, user descriptor
        g0[1] = (uint32_t)(uintptr_t)wsm;             // lds_addr (byte offset in LDS)
        g0[2] = (uint32_t)gaddr;                      // global_addr[31:0]
        g0[3] = ((uint32_t)(gaddr >> 32) & 0x01FFFFFFu) | 0x80000000u; // addr[56:32] | type=2
        v8u g1;
        g1[0] = 0x00020000u;  // workgroup_mask=0, data_size=4B, no flags
        g1[1] = 0x30000000u;  // tensor_dim0 = 12288 (low16 in [31:16])
        g1[2] = 0x00010000u;  // tensor_dim0 hi=0, tensor_dim1 = 1
        g1[3] = 0x30000000u;  // tensor_dim1 hi=0, tile_dim0 = 12288
        g1[4] = 0x00000001u;  // tile_dim1 = 1, tile_dim2 = 0
        g1[5] = 0x00003000u;  // tensor_dim0_stride = 12288
        g1[6] = 0u;
        g1[7] = 0u;
        asm volatile("tensor_load_to_lds %0, %1" :: "s"(g0), "s"(g1) : "memory");
        __builtin_amdgcn_s_wait_tensorcnt(0);
    }

    // ---- stage gathered rows [n0-1 .. n0+TILE] via async global->LDS copies ----
    {
        const int l16 = tid & 15;
        for (int r = tid >> 4; r < XROWS; r += 16) {
            const int p = n0 + r - 1;
            float* lp = xs + r * XSTR + l16 * 4;
            if (p >= 0 && p < L) {
                const long row = GATHER ? (long)inv[p] : (long)p;
                const float* ga = src + row * 64 + l16 * 4;
                const uint32_t lws = (uint32_t)(uintptr_t)lp;
                asm volatile("global_load_async_to_lds_b128 %0, %1, off"
                             :: "v"(lws), "v"(ga) : "memory");
            } else {
                *(float4*)lp = make_float4(0.f, 0.f, 0.f, 0.f);  // SAME padding
            }
        }
        asm volatile("s_wait_asynccnt 0x0" ::: "memory");
    }
    __syncthreads();

    // ---- im2col GEMM: [16 pts x 192] x [192 x 64] via V_WMMA_F32_16X16X4_F32 ----
    const int wave = tid >> 5;        // 8 waves (wave32)
    const int lane = tid & 31;
    const int l16  = lane & 15;
    const int hi   = lane >> 4;       // A/B lanes 16-31 carry K+2,K+3

    v8f acc[2][4];
#pragma unroll
    for (int mt = 0; mt < 2; ++mt)
#pragma unroll
        for (int ft = 0; ft < 4; ++ft)
            acc[mt][ft] = (v8f){0.f, 0.f, 0.f, 0.f, 0.f, 0.f, 0.f, 0.f};

    const int mbase = wave * 32;      // 32 points per wave (2 M-tiles)

#pragma unroll 4
    for (int kk = 0; kk < 48; ++kk) { // 48 K-steps of depth 4 over K=192
        const int win = kk >> 4;                      // conv tap 0..2
        const int ch  = ((kk & 15) << 2) + (hi << 1); // channel pair this half-wave

        // A fragments: lane l16 = row M; ds_load_b64 -> (K, K+1)
        v2f a0 = *(const v2f*)(xs + (mbase +  0 + l16 + win) * XSTR + ch);
        v2f a1 = *(const v2f*)(xs + (mbase + 16 + l16 + win) * XSTR + ch);

        // B fragments: rows = channels ch, ch+1; cols = filters ft*16 + l16
        const float* wb = wsm + (win * CIN + ch) * FOUT;
        v2f b[4];
#pragma unroll
        for (int ft = 0; ft < 4; ++ft) {
            b[ft].x = wb[ft * 16 + l16];
            b[ft].y = wb[FOUT + ft * 16 + l16];
        }

#pragma unroll
        for (int ft = 0; ft < 4; ++ft) {
            acc[0][ft] = __builtin_amdgcn_wmma_f32_16x16x4_f32(
                false, a0, false, b[ft], (short)0, acc[0][ft], false, false);
            acc[1][ft] = __builtin_amdgcn_wmma_f32_16x16x4_f32(
                false, a1, false, b[ft], (short)0, acc[1][ft], false, false);
        }
    }

    // ---- epilogue: bias + scatter (C/D layout: VGPR j -> M = j + hi*8, N = l16) ----
#pragma unroll
    for (int mt = 0; mt < 2; ++mt) {
#pragma unroll
        for (int ft = 0; ft < 4; ++ft) {
            const int f = ft * 16 + l16;
            const float bv = bias[f];
#pragma unroll
            for (int j = 0; j < 8; ++j) {
                int p = n0 + mbase + mt * 16 + j + hi * 8;
                if (p < L) {
                    float val = acc[mt][ft][j] + bv;
                    if (SCATTER) {
                        atomicAdd(dst + (long)segid[p] * 64 + f, val);
                    } else {
                        dst[(long)p * 64 + f] = val;
                    }
                }
            }
        }
    }
}

__global__ __launch_bounds__(256) void zero_f4(float* __restrict__ p, long n4) {
    long i = blockIdx.x * 256L + threadIdx.x;
    if (i < n4) *(float4*)(p + i * 4) = make_float4(0.f, 0.f, 0.f, 0.f);
}

// Y[p] (+)= T[segid[p]]   (float4 lanes; 16 float4 per row)
__global__ __launch_bounds__(256) void gather_add(
    const float* __restrict__ T, const int* __restrict__ segid,
    float* __restrict__ Y, int n, int accumulate)
{
    long t  = blockIdx.x * 256L + threadIdx.x;
    int  p  = (int)(t >> 4);
    int  c4 = (int)(t & 15) * 4;
    if (p < n) {
        float4 v = *(const float4*)(T + (long)segid[p] * 64 + c4);
        float4* yp = (float4*)(Y + (long)p * 64 + c4);
        if (accumulate) {
            float4 o = *yp;
            v.x += o.x; v.y += o.y; v.z += o.z; v.w += o.w;
        }
        *yp = v;
    }
}

extern "C" void kernel_launch(void* const* d_in, const int* in_sizes, int n_in,
                              void* d_out, int out_size, void* d_ws, size_t ws_size,
                              hipStream_t stream) {
    (void)in_sizes; (void)n_in; (void)out_size; (void)ws_size;
    const float* X      = (const float*)d_in[0];   // [N,64]
    const float* pre_w  = (const float*)d_in[1];   // [3,3,64,64]
    const float* pre_b  = (const float*)d_in[2];   // [3,64]
    const float* post_w = (const float*)d_in[3];   // [3,3,64,64]
    const float* post_b = (const float*)d_in[4];   // [3,64]
    const int*   index  = (const int*)d_in[5];     // [3,N]
    const int*   inv    = (const int*)d_in[6];     // [3,N]
    float* Y = (float*)d_out;                      // [N,64]

    float* S = (float*)d_ws;                       // [NSEGS,64] segment sums
    float* T = S + (long)NSEGS * 64;               // [NSEGS,64] post-conv

    const int preBlocks  = (NPTS + TILE - 1) / TILE;
    const int postBlocks = (NSEGS + TILE - 1) / TILE;
    const long seg4      = (long)NSEGS * 64 / 4;
    const int zeroBlocks = (int)((seg4 + 255) / 256);
    const int gaBlocks   = (int)(((long)NPTS * 16 + 255) / 256);

    for (int d = 0; d < 3; ++d) {
        const float* pw = pre_w  + (long)d * KW * CIN * FOUT;
        const float* pb = pre_b  + (long)d * FOUT;
        const float* qw = post_w + (long)d * KW * FOUT * FOUT;
        const float* qb = post_b + (long)d * FOUT;
        const int*   id = index  + (long)d * NPTS;
        const int*   iv = inv    + (long)d * NPTS;

        zero_f4<<<zeroBlocks, 256, 0, stream>>>(S, seg4);

        beam_conv_wmma<true, true><<<preBlocks, 256, SMEMB, stream>>>(
            X, pw, pb, iv, id, S, NPTS);

        beam_conv_wmma<false, false><<<postBlocks, 256, SMEMB, stream>>>(
            S, qw, qb, nullptr, nullptr, T, NSEGS);

        gather_add<<<gaBlocks, 256, 0, stream>>>(T, id, Y, NPTS, d > 0 ? 1 : 0);
    }
}